// GNNpool_67413806678385
// MI455X (gfx1250) — compile-verified
//
#include <hip/hip_runtime.h>
#include <math.h>

typedef float v2f __attribute__((ext_vector_type(2)));
typedef float v8f __attribute__((ext_vector_type(8)));
typedef int   v4i __attribute__((vector_size(16)));

#if __has_builtin(__builtin_amdgcn_global_load_async_to_lds_b128)
#define HAVE_ASYNC_LDS 1
#endif

// 16-byte global->LDS copy; async DMA path (ASYNCcnt) when available.
__device__ __forceinline__ void copy16_g2l(const float* gp, float* lp) {
#ifdef HAVE_ASYNC_LDS
    __builtin_amdgcn_global_load_async_to_lds_b128(
        (__attribute__((address_space(1))) v4i*)(v4i*)(float*)gp,
        (__attribute__((address_space(3))) v4i*)(v4i*)lp,
        0, 0);
#else
    *(float4*)lp = *(const float4*)gp;
#endif
}

__device__ __forceinline__ void wait_async_copies() {
#ifdef HAVE_ASYNC_LDS
#if __has_builtin(__builtin_amdgcn_s_wait_asynccnt)
    __builtin_amdgcn_s_wait_asynccnt(0);
#else
    asm volatile("s_wait_asynccnt 0x0" ::: "memory");
#endif
#endif
}

// ---------------------------------------------------------------------------
// utility: fill a float buffer
// ---------------------------------------------------------------------------
__global__ void fill_kernel(float* __restrict__ p, float v, int n) {
    int i = blockIdx.x * blockDim.x + threadIdx.x;
    int stride = gridDim.x * blockDim.x;
    for (; i < n; i += stride) p[i] = v;
}

// ---------------------------------------------------------------------------
// degree accumulation: deg[dst[e]] += 1  (deg pre-filled with 1.0 = self loop)
// ---------------------------------------------------------------------------
__global__ void deg_kernel(const int* __restrict__ dst, float* __restrict__ deg, int E) {
    int i = blockIdx.x * blockDim.x + threadIdx.x;
    if (i < E) unsafeAtomicAdd(&deg[dst[i]], 1.0f);
}

__global__ void rsqrt_kernel(float* __restrict__ d, int n) {
    int i = blockIdx.x * blockDim.x + threadIdx.x;
    if (i < n) d[i] = rsqrtf(d[i]);
}

// ---------------------------------------------------------------------------
// WMMA GEMM: C[nrows,128] = A[nrows,K] @ W[K,128]
// Block = 256 threads = 8 waves; each wave computes 32 rows x 128 cols
// (2 M-tiles x 8 N-tiles = 16 v8f accumulators).
// Weight panel staged K-chunk by K-chunk (KC=32) into LDS with the CDNA5
// async global->LDS copy (ASYNCcnt); B fragments then come from ds_load.
// LDS row stride 144 floats: lane halves (rows k, k+2) land on disjoint
// bank ranges (2*144 mod 64 == 32) -> conflict-free ds_load_b32.
// Fragment layouts per CDNA5 ISA 7.12.2 (V_WMMA_F32_16X16X4_F32):
//   A 16x4 : lane&15 = M row, lane>>4 selects K pair {0,1}/{2,3}
//   B 4x16 : lane&15 = N col, lane>>4 selects K pair
//   C      : VGPR r -> M = r + 8*(lane>>4), N = lane&15
// ---------------------------------------------------------------------------
#define KC 32
#define LDS_STRIDE 144

__global__ __launch_bounds__(256)
void gemm128_wmma(const float* __restrict__ A, const float* __restrict__ W,
                  float* __restrict__ C, int nrows, int K) {
    __shared__ float smem[KC * LDS_STRIDE];

    const int tid   = threadIdx.x;
    const int lane  = tid & 31;
    const int wave  = tid >> 5;
    const int mbase = blockIdx.x * 256 + wave * 32;
    const int r15   = lane & 15;
    const int half  = lane >> 4;

    v8f acc0[8] = {};
    v8f acc1[8] = {};

    int arow0 = mbase + r15;
    int arow1 = mbase + 16 + r15;
    if (arow0 >= nrows) arow0 = nrows - 1;   // clamp; tail rows never stored
    if (arow1 >= nrows) arow1 = nrows - 1;
    const float* Arow0 = A + (long)arow0 * K;
    const float* Arow1 = A + (long)arow1 * K;

    for (int kc = 0; kc < K; kc += KC) {
        // ---- stage W[kc:kc+KC, 0:128] into LDS (padded rows) ----
        for (int i = tid; i < KC * 32; i += 256) {       // 16B granules
            const int r = i >> 5;
            const int c = (i & 31) << 2;                 // float column
            copy16_g2l(W + (long)(kc + r) * 128 + c, &smem[r * LDS_STRIDE + c]);
        }
        wait_async_copies();
        __syncthreads();

        // ---- compute on the staged panel ----
        for (int k = 0; k < KC; k += 4) {
            const int ka = k + half * 2;                 // K pair for this half
            v2f a0, a1;
            a0.x = Arow0[kc + ka];
            a0.y = Arow0[kc + ka + 1];
            a1.x = Arow1[kc + ka];
            a1.y = Arow1[kc + ka + 1];
#pragma unroll
            for (int t = 0; t < 8; ++t) {
                const int n = t * 16 + r15;
                v2f b;
                b.x = smem[ka * LDS_STRIDE + n];
                b.y = smem[(ka + 1) * LDS_STRIDE + n];
                acc0[t] = __builtin_amdgcn_wmma_f32_16x16x4_f32(
                    false, a0, false, b, (short)0, acc0[t], false, false);
                acc1[t] = __builtin_amdgcn_wmma_f32_16x16x4_f32(
                    false, a1, false, b, (short)0, acc1[t], false, false);
            }
        }
        __syncthreads();
    }

#pragma unroll
    for (int t = 0; t < 8; ++t) {
        const int n = t * 16 + r15;
#pragma unroll
        for (int r = 0; r < 8; ++r) {
            const int m0 = mbase + half * 8 + r;
            const int m1 = m0 + 16;
            if (m0 < nrows) C[(long)m0 * 128 + n] = acc0[t][r];
            if (m1 < nrows) C[(long)m1 * 128 + n] = acc1[t][r];
        }
    }
}

// ---------------------------------------------------------------------------
// WMMA GEMM head: C[nrows,16] = A[nrows,128] @ W[128,16] + bias[16]
// One wave computes a 16x16 tile. Block = 256 threads = 8 waves = 128 rows.
// ---------------------------------------------------------------------------
__global__ __launch_bounds__(256)
void gemm16_wmma(const float* __restrict__ A, const float* __restrict__ W,
                 const float* __restrict__ bias, float* __restrict__ C, int nrows) {
    const int lane  = threadIdx.x & 31;
    const int wave  = threadIdx.x >> 5;
    const int mbase = blockIdx.x * 128 + wave * 16;
    const int r15   = lane & 15;
    const int half  = lane >> 4;

    v8f acc = {};
    int arow = mbase + r15;
    if (arow >= nrows) arow = nrows - 1;
    const float* Arow = A + (long)arow * 128;

    for (int k = 0; k < 128; k += 4) {
        const int kk = k + half * 2;
        v2f a;
        a.x = Arow[kk];
        a.y = Arow[kk + 1];
        v2f b;
        b.x = W[kk * 16 + r15];
        b.y = W[(kk + 1) * 16 + r15];
        acc = __builtin_amdgcn_wmma_f32_16x16x4_f32(
            false, a, false, b, (short)0, acc, false, false);
    }

    const float bn = bias[r15];
#pragma unroll
    for (int r = 0; r < 8; ++r) {
        const int m = mbase + half * 8 + r;
        if (m < nrows) C[(long)m * 16 + r15] = acc[r] + bn;
    }
}

// ---------------------------------------------------------------------------
// Edge scatter: out[dst] += feat[src] * dinv[src]*dinv[dst]
// Messages 0..E-1 are edges; E..E+N-1 are self loops.
// One wave32 per message; each lane handles 4 contiguous floats (float4).
// ---------------------------------------------------------------------------
__global__ __launch_bounds__(256)
void scatter_kernel(const float* __restrict__ feat, const int* __restrict__ src,
                    const int* __restrict__ dst, const float* __restrict__ dinv,
                    float* __restrict__ out, int E, int N) {
    const long gwarp = (long)blockIdx.x * 8 + (threadIdx.x >> 5);
    const int  lane  = threadIdx.x & 31;
    const long total = (long)E + (long)N;
    if (gwarp >= total) return;

    int s, d;
    if (gwarp < E) { s = src[gwarp]; d = dst[gwarp]; }
    else           { s = d = (int)(gwarp - E); }

    const float nrm = dinv[s] * dinv[d];
    const float4 v = *(const float4*)(feat + (long)s * 128 + lane * 4);
    float* o = out + (long)d * 128 + lane * 4;
    unsafeAtomicAdd(o + 0, v.x * nrm);
    unsafeAtomicAdd(o + 1, v.y * nrm);
    unsafeAtomicAdd(o + 2, v.z * nrm);
    unsafeAtomicAdd(o + 3, v.w * nrm);
}

// ---------------------------------------------------------------------------
// Fused  out = act(LN(in + bias) * g + be) [+ res]
// One wave per 128-wide row; float4 per lane; wave32 shuffle reductions.
// ---------------------------------------------------------------------------
__device__ __forceinline__ float elu1(float x) {
    return x > 0.0f ? x : (__expf(x) - 1.0f);
}

__global__ __launch_bounds__(256)
void ln_act_kernel(const float* __restrict__ in, const float* __restrict__ bias,
                   const float* __restrict__ g, const float* __restrict__ be,
                   const float* __restrict__ res, float* __restrict__ out,
                   int nrows, int do_elu) {
    const int row  = blockIdx.x * 8 + (threadIdx.x >> 5);
    const int lane = threadIdx.x & 31;
    if (row >= nrows) return;

    const float4 iv = *(const float4*)(in + (long)row * 128 + lane * 4);
    const float4 bv = *(const float4*)(bias + lane * 4);
    float x0 = iv.x + bv.x, x1 = iv.y + bv.y, x2 = iv.z + bv.z, x3 = iv.w + bv.w;

    float s = x0 + x1 + x2 + x3;
    float q = x0 * x0 + x1 * x1 + x2 * x2 + x3 * x3;
#pragma unroll
    for (int off = 16; off > 0; off >>= 1) {
        s += __shfl_xor(s, off, 32);
        q += __shfl_xor(q, off, 32);
    }
    const float mean = s * (1.0f / 128.0f);
    const float var  = q * (1.0f / 128.0f) - mean * mean;
    const float rstd = rsqrtf(var + 1e-5f);

    const float4 gv  = *(const float4*)(g + lane * 4);
    const float4 bev = *(const float4*)(be + lane * 4);
    float y0 = (x0 - mean) * rstd * gv.x + bev.x;
    float y1 = (x1 - mean) * rstd * gv.y + bev.y;
    float y2 = (x2 - mean) * rstd * gv.z + bev.z;
    float y3 = (x3 - mean) * rstd * gv.w + bev.w;

    if (do_elu) { y0 = elu1(y0); y1 = elu1(y1); y2 = elu1(y2); y3 = elu1(y3); }

    if (res) {
        const float4 rv = *(const float4*)(res + (long)row * 128 + lane * 4);
        y0 += rv.x; y1 += rv.y; y2 += rv.z; y3 += rv.w;
    }

    float4 ov; ov.x = y0; ov.y = y1; ov.z = y2; ov.w = y3;
    *(float4*)(out + (long)row * 128 + lane * 4) = ov;
}

// ---------------------------------------------------------------------------
// Row-wise softmax over 16 logits
// ---------------------------------------------------------------------------
__global__ void softmax16_kernel(const float* __restrict__ logits,
                                 float* __restrict__ out, int n) {
    const int i = blockIdx.x * blockDim.x + threadIdx.x;
    if (i >= n) return;
    const float* L = logits + (long)i * 16;
    float m = L[0];
#pragma unroll
    for (int k = 1; k < 16; ++k) m = fmaxf(m, L[k]);
    float e[16], s = 0.0f;
#pragma unroll
    for (int k = 0; k < 16; ++k) { e[k] = __expf(L[k] - m); s += e[k]; }
    const float inv = 1.0f / s;
    float* O = out + (long)i * 16;
#pragma unroll
    for (int k = 0; k < 16; ++k) O[k] = e[k] * inv;
}

// ---------------------------------------------------------------------------
// launch
// ---------------------------------------------------------------------------
extern "C" void kernel_launch(void* const* d_in, const int* in_sizes, int n_in,
                              void* d_out, int out_size, void* d_ws, size_t ws_size,
                              hipStream_t stream) {
    const float* x   = (const float*)d_in[0];
    const int*   ei  = (const int*)d_in[1];
    const float* W1  = (const float*)d_in[2];
    const float* b1  = (const float*)d_in[3];
    const float* g1  = (const float*)d_in[4];
    const float* be1 = (const float*)d_in[5];
    const float* W2  = (const float*)d_in[6];
    const float* b2  = (const float*)d_in[7];
    const float* g2  = (const float*)d_in[8];
    const float* be2 = (const float*)d_in[9];
    const float* rW  = (const float*)d_in[10];
    const float* rb  = (const float*)d_in[11];
    const float* rg  = (const float*)d_in[12];
    const float* rbe = (const float*)d_in[13];
    const float* mW1 = (const float*)d_in[14];
    const float* mb1 = (const float*)d_in[15];
    const float* mg  = (const float*)d_in[16];
    const float* mbe = (const float*)d_in[17];
    const float* mW2 = (const float*)d_in[18];
    const float* mb2 = (const float*)d_in[19];

    const int N = in_sizes[0] / 384;
    const int E = in_sizes[1] / 2;
    const int* srcI = ei;
    const int* dstI = ei + E;

    float* ws   = (float*)d_ws;
    float* dinv = ws;                          // [N]
    float* T0   = ws + N;                      // [N,128]
    float* T1   = T0 + (size_t)N * 128;        // [N,128]
    float* T2   = T1 + (size_t)N * 128;        // [N,128]
    float* T3   = T2 + (size_t)N * 128;        // [N,128]

    const int gemmBlocks = (N + 255) / 256;
    const int lnBlocks   = (N + 7) / 8;
    const long totalMsgs = (long)E + (long)N;
    const int  scBlocks  = (int)((totalMsgs + 7) / 8);

    // symmetric GCN normalization: deg (with self loop) -> dinv
    fill_kernel<<<256, 256, 0, stream>>>(dinv, 1.0f, N);
    deg_kernel<<<(E + 255) / 256, 256, 0, stream>>>(dstI, dinv, E);
    rsqrt_kernel<<<(N + 255) / 256, 256, 0, stream>>>(dinv, N);

    // layer 1: T0 = x@W1, T1 = x@rW
    gemm128_wmma<<<gemmBlocks, 256, 0, stream>>>(x, W1, T0, N, 384);
    gemm128_wmma<<<gemmBlocks, 256, 0, stream>>>(x, rW, T1, N, 384);

    // aggregate conv1 into T2
    fill_kernel<<<1024, 256, 0, stream>>>(T2, 0.0f, N * 128);
    scatter_kernel<<<scBlocks, 256, 0, stream>>>(T0, srcI, dstI, dinv, T2, E, N);

    // residual branch: T1 = LN(x@rW + rb)  (in-place, no activation)
    ln_act_kernel<<<lnBlocks, 256, 0, stream>>>(T1, rb, rg, rbe, nullptr, T1, N, 0);
    // h (T3) = elu(LN(agg1 + b1)) + T1
    ln_act_kernel<<<lnBlocks, 256, 0, stream>>>(T2, b1, g1, be1, T1, T3, N, 1);

    // layer 2: T0 = h@W2 ; aggregate into T1 ; h2 (T2) = elu(LN(agg2+b2)) + h
    gemm128_wmma<<<gemmBlocks, 256, 0, stream>>>(T3, W2, T0, N, 128);
    fill_kernel<<<1024, 256, 0, stream>>>(T1, 0.0f, N * 128);
    scatter_kernel<<<scBlocks, 256, 0, stream>>>(T0, srcI, dstI, dinv, T1, E, N);
    ln_act_kernel<<<lnBlocks, 256, 0, stream>>>(T1, b2, g2, be2, T3, T2, N, 1);

    // MLP head: T0 = h2@mW1 ; z (T1) = elu(LN(T0+mb1))
    gemm128_wmma<<<gemmBlocks, 256, 0, stream>>>(T2, mW1, T0, N, 128);
    ln_act_kernel<<<lnBlocks, 256, 0, stream>>>(T0, mb1, mg, mbe, nullptr, T1, N, 1);

    // logits (T2, first N*16) = z@mW2 + mb2 ; softmax -> d_out
    gemm16_wmma<<<(N + 127) / 128, 256, 0, stream>>>(T1, mW2, mb2, T2, N);
    softmax16_kernel<<<(N + 255) / 256, 256, 0, stream>>>(T2, (float*)d_out, N);
}